// GCN_64424509440202
// MI455X (gfx1250) — compile-verified
//
#include <hip/hip_runtime.h>
#include <math.h>

typedef __attribute__((ext_vector_type(2))) float v2f;
typedef __attribute__((ext_vector_type(4))) float v4f;
typedef __attribute__((ext_vector_type(8))) float v8f;

// ---------------------------------------------------------------------------
// Fill an int buffer (workspace is poisoned, must zero explicitly).
// ---------------------------------------------------------------------------
__global__ void gcn_fill_i32(int* __restrict__ p, int v, int n) {
    int i = blockIdx.x * blockDim.x + threadIdx.x;
    if (i < n) p[i] = v;
}

// ---------------------------------------------------------------------------
// In-degree histogram: cnt[dst[e]] += 1  (one-time CSR build, int atomics)
// ---------------------------------------------------------------------------
__global__ void gcn_count(const int* __restrict__ dst, int* __restrict__ cnt, int E) {
    int e = blockIdx.x * blockDim.x + threadIdx.x;
    if (e < E) atomicAdd(&cnt[dst[e]], 1);
}

// ---------------------------------------------------------------------------
// Two-level exclusive scan of cnt[] -> rowstart[] (CSR row pointers).
// Stage 1: per-block (256) sums into partial[].
// ---------------------------------------------------------------------------
__global__ void gcn_scan_blocksum(const int* __restrict__ cnt, int* __restrict__ partial, int n) {
    __shared__ int sm[256];
    int t = threadIdx.x;
    int i = blockIdx.x * 256 + t;
    sm[t] = (i < n) ? cnt[i] : 0;
    __syncthreads();
    for (int off = 128; off > 0; off >>= 1) {
        if (t < off) sm[t] += sm[t + off];
        __syncthreads();
    }
    if (t == 0) partial[blockIdx.x] = sm[0];
}

// Stage 2: single-block exclusive scan of partial[] (nb <= 1024).
__global__ void gcn_scan_partials(int* __restrict__ partial, int nb) {
    __shared__ int sm[1024];
    int t = threadIdx.x;
    sm[t] = (t < nb) ? partial[t] : 0;
    __syncthreads();
    for (int off = 1; off < 1024; off <<= 1) {
        int v = sm[t];
        if (t >= off) v += sm[t - off];
        __syncthreads();
        sm[t] = v;
        __syncthreads();
    }
    int excl = (t == 0) ? 0 : sm[t - 1];
    if (t < nb) partial[t] = excl;
}

// Stage 3: per-block exclusive scan + partial offset -> rowstart;
// also fuse dinv[i] = rsqrt(deg_i + 1)  (self-loop).
__global__ void gcn_scan_final(const int* __restrict__ cnt, const int* __restrict__ partial,
                               int* __restrict__ rowstart, float* __restrict__ dinv, int n) {
    __shared__ int sm[256];
    int t = threadIdx.x;
    int i = blockIdx.x * 256 + t;
    int c = (i < n) ? cnt[i] : 0;
    sm[t] = c;
    __syncthreads();
    for (int off = 1; off < 256; off <<= 1) {
        int v = sm[t];
        if (t >= off) v += sm[t - off];
        __syncthreads();
        sm[t] = v;
        __syncthreads();
    }
    if (i < n) {
        int excl = partial[blockIdx.x] + sm[t] - c;   // exclusive = inclusive - self
        rowstart[i] = excl;
        dinv[i] = rsqrtf((float)c + 1.0f);
        if (i == n - 1) rowstart[n] = excl + c;
    }
}

// ---------------------------------------------------------------------------
// Bin edges into CSR slots, storing a packed record {src, norm} per slot.
// norm = dinv[src] * dinv[dst] is computed ONCE here and reused by both
// layers, so the per-layer hot loop needs a single b64 broadcast per edge.
// ---------------------------------------------------------------------------
__global__ void gcn_bin(const int* __restrict__ src, const int* __restrict__ dst,
                        const int* __restrict__ rowstart, const float* __restrict__ dinv,
                        int* __restrict__ cursor, int2* __restrict__ packed, int E) {
    int e = blockIdx.x * blockDim.x + threadIdx.x;
    if (e >= E) return;
    int t = dst[e];
    int s = src[e];
    int slot = rowstart[t] + atomicAdd(&cursor[t], 1);
    float norm = dinv[s] * dinv[t];
    packed[slot] = make_int2(s, __float_as_int(norm));
}

// ---------------------------------------------------------------------------
// C[M,N] = A[M,K] @ B[K,N] (+ bias[N] if bias != nullptr)
// One wave32 computes one 16x16 output tile with V_WMMA_F32_16X16X4_F32.
// A-layout (16x4): lane l holds row m = l&15, K-pair kh = (l>>4)*2.
// B-layout (4x16): lane l holds col n = l&15, same K-pair striping.
// C/D-layout:      VGPR v -> row v + (l>=16 ? 8 : 0), col l&15.
// Tile guard is wave-uniform -> EXEC all-1s for every WMMA.
// ---------------------------------------------------------------------------
__global__ void gcn_gemm_wmma(const float* __restrict__ A, const float* __restrict__ B,
                              const float* __restrict__ bias, float* __restrict__ C,
                              int M, int K, int N) {
    int wave = (int)((blockIdx.x * blockDim.x + threadIdx.x) >> 5);
    int lane = threadIdx.x & 31;
    int tilesN = N >> 4;
    int tm = wave / tilesN;
    int tn = wave - tm * tilesN;
    if (tm * 16 >= M) return;                 // uniform across the wave

    const int mn = lane & 15;                 // row (for A) / col (for B, C)
    const int kh = (lane >> 4) << 1;          // 0 for lanes 0-15, 2 for 16-31

    const float* arow = A + (size_t)(tm * 16 + mn) * K;
    const float* bcol = B + (tn * 16 + mn);

    v8f acc = {0.f, 0.f, 0.f, 0.f, 0.f, 0.f, 0.f, 0.f};
    for (int k = 0; k < K; k += 4) {
        v2f a, b;
        a.x = arow[k + kh];
        a.y = arow[k + kh + 1];
        b.x = bcol[(size_t)(k + kh) * N];
        b.y = bcol[(size_t)(k + kh + 1) * N];
        acc = __builtin_amdgcn_wmma_f32_16x16x4_f32(
            /*neg_a=*/false, a, /*neg_b=*/false, b,
            /*c_mod=*/(short)0, acc, /*reuse_a=*/false, /*reuse_b=*/false);
    }

    int row0 = tm * 16 + ((lane >> 4) << 3);
    int col  = tn * 16 + mn;
    float badd = bias ? bias[col] : 0.0f;
#pragma unroll
    for (int v = 0; v < 8; ++v)
        C[(size_t)(row0 + v) * N + col] = acc[v] + badd;
}

// ---------------------------------------------------------------------------
// Fused CSR aggregation + self-loop + bias + ReLU (one wave32 per node).
//   h[i,:] = relu( sum_{e: dst=i} norm_e * xw[src_e,:] + dinv[i]^2*xw[i,:] + b )
// Lane owns dims (2*lane, 2*lane+1): each neighbor row is ONE coalesced b64
// per lane (256B/wave), plus ONE broadcast b64 for the packed edge record.
// Zero atomics, deterministic accumulation order.
// ---------------------------------------------------------------------------
__global__ void gcn_aggregate(const float* __restrict__ xw, const int2* __restrict__ packed,
                              const int* __restrict__ rowstart, const float* __restrict__ dinv,
                              const float* __restrict__ bias, float* __restrict__ h, int n) {
    int wave = (int)((blockIdx.x * blockDim.x + threadIdx.x) >> 5);
    int lane = threadIdx.x & 31;
    if (wave >= n) return;
    int i = wave;
    int d2 = lane << 1;                                   // dims d2, d2+1

    float di = dinv[i];
    float sl = di * di;                                   // self-loop weight
    v2f r = *(const v2f*)(xw + (size_t)i * 64 + d2);
    v2f acc;
    acc.x = r.x * sl;
    acc.y = r.y * sl;

    int beg = rowstart[i];
    int end = rowstart[i + 1];
    for (int j = beg; j < end; ++j) {
        int2 p = packed[j];                               // same addr all lanes -> broadcast
        int s = p.x;
        float norm = __int_as_float(p.y);
        v2f sr = *(const v2f*)(xw + (size_t)s * 64 + d2); // one b64 per lane, coalesced
        acc.x += sr.x * norm;
        acc.y += sr.y * norm;
    }

    v2f bb = *(const v2f*)(bias + d2);
    v2f out;
    out.x = acc.x + bb.x;
    out.y = acc.y + bb.y;
    out.x = out.x > 0.f ? out.x : 0.f;
    out.y = out.y > 0.f ? out.y : 0.f;
    *(v2f*)(h + (size_t)i * 64 + d2) = out;
}

// ---------------------------------------------------------------------------
// Row softmax over 16 logits per node (b128 loads/stores).
// ---------------------------------------------------------------------------
__global__ void gcn_softmax16(const float* __restrict__ logits, float* __restrict__ out, int n) {
    int i = blockIdx.x * blockDim.x + threadIdx.x;
    if (i >= n) return;
    v4f v[4];
#pragma unroll
    for (int q = 0; q < 4; ++q)
        v[q] = *(const v4f*)(logits + (size_t)i * 16 + q * 4);
    float mx = -INFINITY;
#pragma unroll
    for (int q = 0; q < 4; ++q)
#pragma unroll
        for (int j = 0; j < 4; ++j) mx = fmaxf(mx, v[q][j]);
    float s = 0.f;
#pragma unroll
    for (int q = 0; q < 4; ++q)
#pragma unroll
        for (int j = 0; j < 4; ++j) { v[q][j] = expf(v[q][j] - mx); s += v[q][j]; }
    float inv = 1.0f / s;
#pragma unroll
    for (int q = 0; q < 4; ++q) {
        v4f o;
#pragma unroll
        for (int j = 0; j < 4; ++j) o[j] = v[q][j] * inv;
        *(v4f*)(out + (size_t)i * 16 + q * 4) = o;
    }
}

// ---------------------------------------------------------------------------
// Orchestration
// ---------------------------------------------------------------------------
static inline int cdiv(long long a, long long b) { return (int)((a + b - 1) / b); }

extern "C" void kernel_launch(void* const* d_in, const int* in_sizes, int n_in,
                              void* d_out, int out_size, void* d_ws, size_t ws_size,
                              hipStream_t stream) {
    const float* x   = (const float*)d_in[0];
    const float* W1  = (const float*)d_in[1];
    const float* b1  = (const float*)d_in[2];
    const float* W2  = (const float*)d_in[3];
    const float* b2  = (const float*)d_in[4];
    const float* Wl  = (const float*)d_in[5];
    const float* bl  = (const float*)d_in[6];
    const int*  eidx = (const int*)d_in[7];

    const int DIM_IN = 128, H1 = 64, DOUT = 16;
    const int n = in_sizes[0] / DIM_IN;       // 100000
    const int E = in_sizes[7] / 2;            // 1600000
    const int* src = eidx;
    const int* dst = eidx + E;

    const int nb = cdiv(n, 256);              // scan blocks (<= 1024 required)

    // Workspace carve-up (16B-aligned sections).
    char* w = (char*)d_ws;
    size_t off = 0;
    auto take = [&](size_t bytes) {
        char* p = w + off;
        off += (bytes + 15) & ~(size_t)15;
        return p;
    };
    float* dinv     = (float*)take((size_t)n * 4);
    int*   cnt      = (int*)  take((size_t)n * 4);
    int*   rowstart = (int*)  take((size_t)(n + 1) * 4);
    int*   cursor   = (int*)  take((size_t)n * 4);
    int2*  packed   = (int2*) take((size_t)E * 8);        // {src, norm} per CSR slot
    int*   partial  = (int*)  take((size_t)1024 * 4);
    float* buf0     = (float*)take((size_t)n * 64 * 4);   // xW
    float* buf1     = (float*)take((size_t)n * 64 * 4);   // h

    float* logits = (float*)d_out;
    float* probs  = logits + (size_t)n * DOUT;

    const int B = 256;

    // --- CSR build (one-time): counts -> scan -> bin(+norm); fused dinv --
    gcn_fill_i32<<<cdiv(n, B), B, 0, stream>>>(cnt, 0, n);
    gcn_count<<<cdiv(E, B), B, 0, stream>>>(dst, cnt, E);
    gcn_scan_blocksum<<<nb, 256, 0, stream>>>(cnt, partial, n);
    gcn_scan_partials<<<1, 1024, 0, stream>>>(partial, nb);
    gcn_scan_final<<<nb, 256, 0, stream>>>(cnt, partial, rowstart, dinv, n);
    gcn_fill_i32<<<cdiv(n, B), B, 0, stream>>>(cursor, 0, n);
    gcn_bin<<<cdiv(E, B), B, 0, stream>>>(src, dst, rowstart, dinv, cursor, packed, E);

    // --- layer 1: xw1 = x @ W1 ; h1 = relu(Â xw1 + b1) -------------------
    {
        int waves = (n / 16) * (H1 / 16);
        gcn_gemm_wmma<<<cdiv((long long)waves * 32, B), B, 0, stream>>>(
            x, W1, nullptr, buf0, n, DIM_IN, H1);
    }
    gcn_aggregate<<<cdiv((long long)n * 32, B), B, 0, stream>>>(
        buf0, packed, rowstart, dinv, b1, buf1, n);

    // --- layer 2: xw2 = h1 @ W2 ; h2 = relu(Â xw2 + b2) ------------------
    {
        int waves = (n / 16) * (64 / 16);
        gcn_gemm_wmma<<<cdiv((long long)waves * 32, B), B, 0, stream>>>(
            buf1, W2, nullptr, buf0, n, 64, 64);
    }
    gcn_aggregate<<<cdiv((long long)n * 32, B), B, 0, stream>>>(
        buf0, packed, rowstart, dinv, b2, buf1, n);

    // --- head: logits = h2 @ Wl + bl ; softmax ---------------------------
    {
        int waves = (n / 16) * (DOUT / 16);
        gcn_gemm_wmma<<<cdiv((long long)waves * 32, B), B, 0, stream>>>(
            buf1, Wl, bl, logits, n, 64, DOUT);
    }
    gcn_softmax16<<<cdiv(n, B), B, 0, stream>>>(logits, probs, n);
}